// KinematicChain_Numpy_57011395887803
// MI455X (gfx1250) — compile-verified
//
#include <hip/hip_runtime.h>
#include <math.h>

typedef float v2f __attribute__((ext_vector_type(2)));
typedef float v4f __attribute__((ext_vector_type(4)));
typedef float v8f __attribute__((ext_vector_type(8)));

#define BATCH_PER_BLOCK 32
#define THREADS 256

// SMPL parent table (compile-time for the unrolled chain)
#define PAR_LIST {0,0,0,0,1,2,3,4,5,6,7,8,9,9,9,12,13,14,16,17,18,19,20,21}

__global__ __launch_bounds__(THREADS)
void fk_chain_kernel(const float* __restrict__ theta,
                     const float* __restrict__ rest,
                     const float* __restrict__ bonef,
                     const int*   __restrict__ parents,
                     float* __restrict__ kp_out,
                     float* __restrict__ orient_out,
                     float* __restrict__ l2w_out,
                     int Btot)
{
    // One 64B slot per (local batch, joint). Phase 1 writes rel^T (row-major);
    // phase 2 overwrites each slot with that joint's l2w (row-major) after use.
    __shared__ __align__(16) float lds[BATCH_PER_BLOCK * 24 * 16];

    const int tid       = threadIdx.x;
    const int blockBase = blockIdx.x * BATCH_PER_BLOCK;

    // ---------------- Phase 1: Rodrigues + relative transforms -> LDS ----------------
    for (int idx = tid; idx < BATCH_PER_BLOCK * 24; idx += THREADS) {
        int bl = idx / 24;
        int j  = idx - bl * 24;
        int batch = blockBase + bl;
        if (batch >= Btot) batch = Btot - 1;   // clamp: results unused, keeps loads valid

        const float* th = theta + ((size_t)batch * 24 + j) * 3;
        float tx = th[0], ty = th[1], tz = th[2];

        // angle = || axisang + 1e-8 ||  (per-component offset, matches reference)
        float ax = tx + 1e-8f, ay = ty + 1e-8f, az = tz + 1e-8f;
        float ang  = sqrtf(ax * ax + ay * ay + az * az);
        float inva = 1.0f / ang;
        float half = 0.5f * ang;
        float s    = sinf(half) * inva;        // sin(half) * axis, axis = t/ang
        float qw = cosf(half);
        float qx = s * tx, qy = s * ty, qz = s * tz;
        float qn = 1.0f / sqrtf(qw * qw + qx * qx + qy * qy + qz * qz);
        qw *= qn; qx *= qn; qy *= qn; qz *= qn;

        float w2 = qw * qw, x2 = qx * qx, y2 = qy * qy, z2 = qz * qz;
        float wx = qw * qx, wy = qw * qy, wz = qw * qz;
        float xy = qx * qy, xz = qx * qz, yz = qy * qz;

        float R00 = w2 + x2 - y2 - z2, R01 = 2.f*xy - 2.f*wz, R02 = 2.f*wy + 2.f*xz;
        float R10 = 2.f*wz + 2.f*xy, R11 = w2 - x2 + y2 - z2, R12 = 2.f*yz - 2.f*wx;
        float R20 = 2.f*xz - 2.f*wy, R21 = 2.f*wx + 2.f*yz, R22 = w2 - x2 - y2 + z2;

        const float* rp = rest + ((size_t)batch * 24 + j) * 3;
        float ttx, tty, ttz;
        if (j == 0) {
            ttx = rp[0]; tty = rp[1]; ttz = rp[2];
        } else {
            int p = parents[j];
            const float* rpp = rest + ((size_t)batch * 24 + p) * 3;
            float bf = bonef[j - 1];
            float sc = sqrtf(bf * bf + 1e-36f);
            ttx = (rp[0] - rpp[0]) * sc;
            tty = (rp[1] - rpp[1]) * sc;
            ttz = (rp[2] - rpp[2]) * sc;
        }

        // Store rel^T row-major: row m = column m of rel.
        float* slot = &lds[idx * 16];
        v4f r;
        r.x = R00; r.y = R10; r.z = R20; r.w = 0.f;  ((v4f*)slot)[0] = r;
        r.x = R01; r.y = R11; r.z = R21; r.w = 0.f;  ((v4f*)slot)[1] = r;
        r.x = R02; r.y = R12; r.z = R22; r.w = 0.f;  ((v4f*)slot)[2] = r;
        r.x = ttx; r.y = tty; r.z = ttz; r.w = 1.f;  ((v4f*)slot)[3] = r;
    }
    __syncthreads();

    // ---------------- Phase 2: chain via V_WMMA_F32_16X16X4_F32 ----------------
    // Wave w handles local batches 4w..4w+3. Block b = batch-in-wave sits at
    // rows/cols 4b..4b+3 of the 16x16 D; D diag block = child^T.
    const int lane = tid & 31;
    const int wave = tid >> 5;
    const int n    = lane & 15;           // matrix row (A) / column (B,D) index
    const int b    = n >> 2;              // which of 4 batches in this wave
    const int ip   = n & 3;               // row/col within the 4x4 block
    const int k0   = (lane < 16) ? 0 : 2; // K half held by this lane (A & B layout)
    const int bl   = wave * 4 + b;
    const int batch = blockBase + bl;
    const bool inrange = (batch < Btot);
    // D layout: rows 0-7 live in lanes 0-15, rows 8-15 in lanes 16-31.
    const bool active = ((lane < 16) == (b < 2));
    const bool odd    = (b & 1) != 0;     // block rows start at D VGPR 4 when odd
    const int aoff    = ip * 4 + k0;      // element offset inside a 16-float slot

    // Hoisted base pointers: every in-loop access becomes base + constant imm.
    float* slotbase = &lds[bl * 24 * 16 ];                              // + j*16 + ...
    float* l2wp = l2w_out    + (size_t)batch * 384 + (size_t)ip * 4;    // + j*16
    float* orp  = orient_out + (size_t)batch * 216 + (size_t)ip * 3;    // + j*9
    float* kpp  = kp_out     + (size_t)batch * 72  + ip;                // + j*3

    // Root's "parent" = identity, built directly in B layout.
    v2f Bm;
    Bm.x = (ip == k0)       ? 1.0f : 0.0f;
    Bm.y = (ip == (k0 + 1)) ? 1.0f : 0.0f;

    constexpr int PARC[24] = PAR_LIST;

#pragma unroll
    for (int j = 0; j < 24; ++j) {
        if (j > 0) {
            // B = parent l2w^T: lane n needs parent[ip][k0], parent[ip][k0+1]
            // (row-major slot -> one b64 load, immediate offset).
            Bm = *(const v2f*)(slotbase + PARC[j] * 16 + aoff);
        }
        // A = rel^T in A layout: lane (m = 4b+ip) needs rel^T[ip][k0..k0+1].
        v2f A = *(const v2f*)(slotbase + j * 16 + aoff);

        v8f D = {};
        D = __builtin_amdgcn_wmma_f32_16x16x4_f32(
                /*neg_a=*/false, A, /*neg_b=*/false, Bm,
                /*c_mod=*/(short)0, D, /*reuse_a=*/false, /*reuse_b=*/false);

        // Active lane holds child row `ip` of its batch in D[0..3] or D[4..7]
        // (constant indices + one 2-way select; avoids dynamic-index cndmask tree).
        float r0 = odd ? D[4] : D[0];
        float r1 = odd ? D[5] : D[1];
        float r2 = odd ? D[6] : D[2];
        float r3 = odd ? D[7] : D[3];

        if (active) {
            // Keep l2w in LDS (row-major) for descendants' B loads.
            v4f row; row.x = r0; row.y = r1; row.z = r2; row.w = r3;
            *(v4f*)(slotbase + j * 16 + ip * 4) = row;
            if (inrange) {
                *(v4f*)(l2wp + j * 16) = row;
                if (ip < 3) {
                    orp[j * 9 + 0] = r0;
                    orp[j * 9 + 1] = r1;
                    orp[j * 9 + 2] = r2;
                    kpp[j * 3] = r3;      // translation = child[ip][3]
                }
            }
        }
    }
}

extern "C" void kernel_launch(void* const* d_in, const int* in_sizes, int n_in,
                              void* d_out, int out_size, void* d_ws, size_t ws_size,
                              hipStream_t stream) {
    const float* theta   = (const float*)d_in[0];
    const float* rest    = (const float*)d_in[1];
    const float* bonef   = (const float*)d_in[2];
    const int*   parents = (const int*)d_in[3];

    int Btot = in_sizes[0] / 72;  // theta is (B, 24, 3)

    float* kp     = (float*)d_out;                 // B*24*3
    float* orient = kp + (size_t)Btot * 72;        // B*24*9
    float* l2w    = kp + (size_t)Btot * 288;       // B*24*16

    int grid = (Btot + BATCH_PER_BLOCK - 1) / BATCH_PER_BLOCK;
    fk_chain_kernel<<<dim3(grid), dim3(THREADS), 0, stream>>>(
        theta, rest, bonef, parents, kp, orient, l2w, Btot);
}